// RippleNet_31619549233423
// MI455X (gfx1250) — compile-verified
//
#include <hip/hip_runtime.h>
#include <hip/hip_bf16.h>

#define DIM      20
#define HOP      3
#define MEM      32
#define NB       8192
#define NREL     32
#define RELSZ    (DIM * DIM)       // 400
#define QCOLS    (NREL * DIM)      // 640
#define KGE_W    0.01f
#define L2_W     1e-7f

typedef __attribute__((ext_vector_type(2))) float v2f;
typedef __attribute__((ext_vector_type(8))) float v8f;

// ---------------------------------------------------------------------------
// Wave32 reductions (gfx1250 is wave32-only)
// ---------------------------------------------------------------------------
__device__ __forceinline__ float wave_red_add(float v) {
    #pragma unroll
    for (int off = 16; off > 0; off >>= 1) v += __shfl_xor(v, off, 32);
    return v;
}
__device__ __forceinline__ float wave_red_max(float v) {
    #pragma unroll
    for (int off = 16; off > 0; off >>= 1) v = fmaxf(v, __shfl_xor(v, off, 32));
    return v;
}

// ---------------------------------------------------------------------------
// Kernel 1: Q[b, rel*20+j] = sum_i item_e[b,i] * R_rel[i,j]
// Dense GEMM [8192,20] x [20,640] via V_WMMA_F32_16X16X4_F32.
// One wave per 16x16 output tile. K=20 -> 5 chunks of K=4.
//
// A 16x4 f32 layout: lanes 0-15 hold M=0..15 {K=0 in v0, K=1 in v1},
//                    lanes 16-31 hold M=0..15 {K=2, K=3}.
// B 4x16 f32 layout (row-striped): lanes 0-15 hold N=0..15 {K=0,K=1},
//                    lanes 16-31 {K=2,K=3}.
// D 16x16 f32: lanes 0-15: vgpr v -> (M=v,   N=lane),
//              lanes16-31: vgpr v -> (M=v+8, N=lane-16).
// ---------------------------------------------------------------------------
__global__ __launch_bounds__(32) void q_gemm_kernel(
    const int*   __restrict__ items,
    const float* __restrict__ entity_emb,
    const float* __restrict__ relation_emb,
    float*       __restrict__ Q)
{
    const int mb   = blockIdx.x;           // row tile (16 batch rows)
    const int nb   = blockIdx.y;           // col tile (16 of 640 cols)
    const int lane = threadIdx.x;          // 0..31
    const int mrow  = lane & 15;
    const int khalf = lane >> 4;           // 0: K{0,1}, 1: K{2,3}

    const int b    = mb * 16 + mrow;
    const int item = items[b];
    const float* ivec = entity_emb + (long)item * DIM;

    // B-fragment column: n = rel*20 + j  -> element (k, n) = relation_emb[rel*400 + k*20 + j]
    const int n   = nb * 16 + mrow;
    const int rel = n / DIM;
    const int j   = n % DIM;
    const float* Rcol = relation_emb + rel * RELSZ + j;

    v8f c = {};
    #pragma unroll
    for (int kk = 0; kk < 5; ++kk) {
        const int k0 = kk * 4 + khalf * 2;
        v2f a, bf;
        a.x  = ivec[k0];
        a.y  = ivec[k0 + 1];
        bf.x = Rcol[k0 * DIM];
        bf.y = Rcol[(k0 + 1) * DIM];
        c = __builtin_amdgcn_wmma_f32_16x16x4_f32(
                /*neg_a=*/false, a, /*neg_b=*/false, bf,
                /*c_mod=*/(short)0, c, /*reuse_a=*/false, /*reuse_b=*/false);
    }

    const int N     = lane & 15;
    const int Mbase = (lane >> 4) * 8;
    #pragma unroll
    for (int v = 0; v < 8; ++v) {
        const int M = Mbase + v;
        Q[(long)(mb * 16 + M) * QCOLS + nb * 16 + N] = c[v];
    }
}

// ---------------------------------------------------------------------------
// Kernel 2: one block per batch element b. 96 threads = 3 waves (one per hop),
// lane m = memory slot (MEM == wave32). relation_emb staged in LDS with
// stride 401 dwords (401 mod 64 = 17, odd) to break bank-conflict patterns.
// ---------------------------------------------------------------------------
#define RSTRIDE 401

template <bool USE_Q>
__global__ __launch_bounds__(96) void triple_kernel(
    const int*   __restrict__ items,
    const int*   __restrict__ labels,
    const int*   __restrict__ hidx,
    const int*   __restrict__ ridx,
    const int*   __restrict__ tidxv,
    const float* __restrict__ entity_emb,
    const float* __restrict__ relation_emb,
    const float* __restrict__ Q,
    float*       __restrict__ scores,
    float*       __restrict__ bce_part,
    float*       __restrict__ sig_part,
    float*       __restrict__ l2_part)
{
    __shared__ float R_lds[NREL * RSTRIDE];
    __shared__ float item_lds[DIM];
    __shared__ float o_part[HOP][DIM];
    __shared__ float o_sum_lds[DIM];
    __shared__ float sig_wave[HOP];
    __shared__ float l2_wave[HOP];

    const int b   = blockIdx.x;
    const int tid = threadIdx.x;
    const int k   = tid >> 5;   // hop = wave id
    const int m   = tid & 31;   // memory slot = lane

    // Stage all 32 relation matrices into LDS (51.3 KB with padding).
    for (int idx = tid; idx < NREL * RELSZ; idx += 96) {
        const int rel = idx / RELSZ;
        const int c   = idx - rel * RELSZ;
        R_lds[rel * RSTRIDE + c] = relation_emb[idx];
    }
    const int item = items[b];
    if (tid < DIM) item_lds[tid] = entity_emb[(long)item * DIM + tid];
    __syncthreads();

    const int base = (b * HOP + k) * MEM + m;
    const int hid = hidx[base];
    const int rid = ridx[base];
    const int tei = tidxv[base];

    // Entity rows are 80 bytes = 16B aligned -> float4 loads.
    float he[DIM], te[DIM];
    {
        const float4* hp = (const float4*)(entity_emb + (long)hid * DIM);
        const float4* tp = (const float4*)(entity_emb + (long)tei * DIM);
        #pragma unroll
        for (int q4 = 0; q4 < 5; ++q4) {
            const float4 hv = hp[q4];
            const float4 tv = tp[q4];
            he[q4 * 4 + 0] = hv.x; he[q4 * 4 + 1] = hv.y;
            he[q4 * 4 + 2] = hv.z; he[q4 * 4 + 3] = hv.w;
            te[q4 * 4 + 0] = tv.x; te[q4 * 4 + 1] = tv.y;
            te[q4 * 4 + 2] = tv.z; te[q4 * 4 + 3] = tv.w;
        }
    }

    const float* Rrow = &R_lds[rid * RSTRIDE];

    float p = 0.f;
    if (USE_Q) {
        const float* qrow = Q + (long)b * QCOLS + rid * DIM;
        #pragma unroll
        for (int jj = 0; jj < DIM; ++jj) p += qrow[jj] * he[jj];
    }

    // Rt = R @ t, hRt = h . Rt; fold ||R||^2 into the same pass.
    // (fallback also computes Rh = R @ h and p = item . Rh)
    float rsq = 0.f, hRt = 0.f;
    for (int i = 0; i < DIM; ++i) {
        float rt = 0.f;
        float rh = 0.f;
        #pragma unroll
        for (int jj = 0; jj < DIM; ++jj) {
            const float Rv = Rrow[i * DIM + jj];
            rt  = fmaf(Rv, te[jj], rt);
            rsq = fmaf(Rv, Rv, rsq);
            if (!USE_Q) rh = fmaf(Rv, he[jj], rh);
        }
        hRt = fmaf(he[i], rt, hRt);
        if (!USE_Q) p = fmaf(item_lds[i], rh, p);
    }

    // Softmax over memory slots = intra-wave reduction (MEM == wave32).
    const float pmax = wave_red_max(p);
    const float e    = expf(p - pmax);
    const float esum = wave_red_add(e);
    const float pn   = e / esum;

    // o_k[i] = sum_m pn * t_e[i]  (per-hop, lane-reduced)
    #pragma unroll
    for (int i = 0; i < DIM; ++i) {
        const float v = wave_red_add(pn * te[i]);
        if (m == 0) o_part[k][i] = v;
    }

    // KGE sigmoid + L2 partials
    const float sig = 1.f / (1.f + expf(-hRt));
    float l2 = rsq;
    #pragma unroll
    for (int i = 0; i < DIM; ++i) l2 = fmaf(he[i], he[i], fmaf(te[i], te[i], l2));

    const float sigw = wave_red_add(sig);
    const float l2w  = wave_red_add(l2);
    if (m == 0) { sig_wave[k] = sigw; l2_wave[k] = l2w; }
    __syncthreads();

    if (tid < DIM) o_sum_lds[tid] = o_part[0][tid] + o_part[1][tid] + o_part[2][tid];
    __syncthreads();

    if (tid == 0) {
        float dot = 0.f;
        #pragma unroll
        for (int i = 0; i < DIM; ++i) dot = fmaf(item_lds[i], o_sum_lds[i], dot);
        const float s = 1.f / (1.f + expf(-dot));
        scores[b] = s;

        const float y   = (float)labels[b];
        const float eps = 1e-12f;
        const float sc  = fminf(fmaxf(s, eps), 1.f - eps);
        bce_part[b] = -(y * logf(sc) + (1.f - y) * logf(1.f - sc));
        sig_part[b] = sig_wave[0] + sig_wave[1] + sig_wave[2];
        l2_part[b]  = l2_wave[0] + l2_wave[1] + l2_wave[2];
    }
}

// ---------------------------------------------------------------------------
// Kernel 3: deterministic fixed-order reduction of per-sample partials -> loss
// ---------------------------------------------------------------------------
__global__ __launch_bounds__(256) void reduce_kernel(
    const float* __restrict__ bce_part,
    const float* __restrict__ sig_part,
    const float* __restrict__ l2_part,
    float*       __restrict__ out_loss)
{
    __shared__ float sb[256], ss[256], sl[256];
    const int tid = threadIdx.x;
    float ab = 0.f, as = 0.f, al = 0.f;
    for (int i = tid; i < NB; i += 256) {
        ab += bce_part[i];
        as += sig_part[i];
        al += l2_part[i];
    }
    sb[tid] = ab; ss[tid] = as; sl[tid] = al;
    __syncthreads();
    for (int st = 128; st > 0; st >>= 1) {
        if (tid < st) {
            sb[tid] += sb[tid + st];
            ss[tid] += ss[tid + st];
            sl[tid] += sl[tid + st];
        }
        __syncthreads();
    }
    if (tid == 0) {
        const float loss = sb[0] / (float)NB
                         - KGE_W * (ss[0] / ((float)NB * (float)MEM))
                         + L2_W * sl[0];
        out_loss[0] = loss;
    }
}

// ---------------------------------------------------------------------------
extern "C" void kernel_launch(void* const* d_in, const int* in_sizes, int n_in,
                              void* d_out, int out_size, void* d_ws, size_t ws_size,
                              hipStream_t stream) {
    const int*   items        = (const int*)d_in[0];
    const int*   labels       = (const int*)d_in[1];
    const int*   h            = (const int*)d_in[2];
    const int*   r            = (const int*)d_in[3];
    const int*   t            = (const int*)d_in[4];
    const float* entity_emb   = (const float*)d_in[5];
    const float* relation_emb = (const float*)d_in[6];

    float* out = (float*)d_out;                 // [0..8191] scores, [8192] loss

    const size_t q_floats       = (size_t)NB * QCOLS;       // 5,242,880
    const size_t partial_floats = (size_t)3 * NB;           // 24,576
    const bool   use_q = ws_size >= (q_floats + partial_floats) * sizeof(float);

    float* Q   = (float*)d_ws;
    float* par = use_q ? ((float*)d_ws + q_floats) : (float*)d_ws;
    float* bce_part = par;
    float* sig_part = par + NB;
    float* l2_part  = par + 2 * NB;

    if (use_q) {
        // WMMA GEMM: Q = ItemE[8192,20] @ Rall[20,640]
        dim3 grid(NB / 16, QCOLS / 16);
        q_gemm_kernel<<<grid, 32, 0, stream>>>(items, entity_emb, relation_emb, Q);
        triple_kernel<true><<<NB, 96, 0, stream>>>(
            items, labels, h, r, t, entity_emb, relation_emb, Q,
            out, bce_part, sig_part, l2_part);
    } else {
        triple_kernel<false><<<NB, 96, 0, stream>>>(
            items, labels, h, r, t, entity_emb, relation_emb, Q,
            out, bce_part, sig_part, l2_part);
    }
    reduce_kernel<<<1, 256, 0, stream>>>(bce_part, sig_part, l2_part, out + NB);
}